// BaseC_59390807769248
// MI455X (gfx1250) — compile-verified
//
#include <hip/hip_runtime.h>
#include <math.h>

// Problem constants (from reference)
#define BB 256
#define LL_TXT 512
#define LA 8
#define LLEFT 64
#define DD 300
#define DD4 75          // DD / 4 float4s per row
#define PP 3

typedef float v2f __attribute__((ext_vector_type(2)));
typedef float v8f __attribute__((ext_vector_type(8)));

// ---------------------------------------------------------------------------
// K0: per-batch lengths + aspect-embedding mean
// ---------------------------------------------------------------------------
__global__ void prep_kernel(const int* __restrict__ text,
                            const int* __restrict__ aspect,
                            const int* __restrict__ left,
                            const float* __restrict__ embed,
                            float* __restrict__ asp_e,
                            float* __restrict__ lens) {
    __shared__ int s_cnt[256];
    __shared__ float s_alen;
    int b = blockIdx.x;
    int tid = threadIdx.x;

    int c = 0;
    for (int i = tid; i < LL_TXT; i += 256) c += (text[b * LL_TXT + i] != 0);
    s_cnt[tid] = c;
    __syncthreads();
    for (int s = 128; s > 0; s >>= 1) {
        if (tid < s) s_cnt[tid] += s_cnt[tid + s];
        __syncthreads();
    }
    if (tid == 0) {
        int a = 0;
        for (int i = 0; i < LA; ++i) a += (aspect[b * LA + i] != 0);
        int ll = 0;
        for (int i = 0; i < LLEFT; ++i) ll += (left[b * LLEFT + i] != 0);
        float alen = (float)a, llen = (float)ll;
        lens[b * 4 + 0] = llen - alen;        // start
        lens[b * 4 + 1] = llen;               // end
        lens[b * 4 + 2] = (float)s_cnt[0];    // mem_len
        lens[b * 4 + 3] = alen;
        s_alen = alen;
    }
    __syncthreads();
    float inva = 1.0f / s_alen;
    for (int d = tid; d < DD; d += 256) {
        float s = 0.0f;
        for (int j = 0; j < LA; ++j)
            s += embed[(long)aspect[b * LA + j] * DD + d];
        asp_e[b * DD + d] = s * inva;
    }
}

// ---------------------------------------------------------------------------
// Kc: cvec = Wk @ w_mlp[:D], d0 = bk . w_mlp[:D]
// ---------------------------------------------------------------------------
__global__ void cvec_kernel(const float* __restrict__ Wk,
                            const float* __restrict__ bk,
                            const float* __restrict__ wmlp,
                            float* __restrict__ cvec_d0) {
    int i = blockIdx.x * blockDim.x + threadIdx.x;
    if (i < DD) {
        float s = 0.0f;
        for (int j = 0; j < DD; ++j) s += Wk[i * DD + j] * wmlp[j];
        cvec_d0[i] = s;
    }
    if (i == DD) {
        float s = 0.0f;
        for (int j = 0; j < DD; ++j) s += bk[j] * wmlp[j];
        cvec_d0[DD] = s;
    }
}

// ---------------------------------------------------------------------------
// K1: per (b,l) gather pass (float4 / global_load_b128):
//     u[b,l] = w_l * (embed_row . cvec) + d0
//     v_s[b,:] = (sum_l embed_row) / mem_len     (unweighted mean)
// one workgroup (8 wave32s) per batch row
// ---------------------------------------------------------------------------
__global__ void pass1_kernel(const int* __restrict__ text,
                             const float* __restrict__ embed,
                             const float* __restrict__ cvec_d0,
                             const float* __restrict__ lens,
                             float* __restrict__ u,
                             float* __restrict__ v_s) {
    __shared__ int    s_text[LL_TXT];
    __shared__ float4 s_cvec[DD4 + 1];
    __shared__ float4 s_part[8][DD4 + 1];
    int b = blockIdx.x;
    int tid = threadIdx.x;
    int wave = tid >> 5, lane = tid & 31;

    for (int i = tid; i < LL_TXT; i += 256) s_text[i] = text[b * LL_TXT + i];
    const float4* cvec4 = (const float4*)cvec_d0;
    for (int i = tid; i < DD4; i += 256) s_cvec[i] = cvec4[i];
    float d0 = cvec_d0[DD];
    float start = lens[b * 4 + 0];
    float endv  = lens[b * 4 + 1];
    float mlen  = lens[b * 4 + 2];
    __syncthreads();

    float4 acc[3];
#pragma unroll
    for (int t = 0; t < 3; ++t) acc[t] = make_float4(0.f, 0.f, 0.f, 0.f);

    for (int l = wave; l < LL_TXT; l += 8) {
        const float4* row4 = (const float4*)(embed + (long)s_text[l] * DD);
        float dot = 0.0f;
#pragma unroll
        for (int t = 0; t < 3; ++t) {
            int i = lane + 32 * t;
            if (i < DD4) {
                float4 v = row4[i];
                float4 cv = s_cvec[i];
                dot += v.x * cv.x + v.y * cv.y + v.z * cv.z + v.w * cv.w;
                acc[t].x += v.x; acc[t].y += v.y; acc[t].z += v.z; acc[t].w += v.w;
            }
        }
        for (int off = 16; off > 0; off >>= 1) dot += __shfl_xor(dot, off, 32);
        if (lane == 0) {
            float fl = (float)l;
            float ldst = fl < start ? (start - fl) : (fl <= endv ? 0.0f : fl - endv);
            float w = 1.0f - ldst / mlen;
            w = (fl < mlen) ? w : 0.0f;
            u[b * LL_TXT + l] = w * dot + d0;
        }
    }
#pragma unroll
    for (int t = 0; t < 3; ++t) {
        int i = lane + 32 * t;
        if (i < DD4) s_part[wave][i] = acc[t];
    }
    __syncthreads();
    float inv = 1.0f / mlen;
    float4* vs4 = (float4*)(v_s + (size_t)b * DD);
    for (int i = tid; i < DD4; i += 256) {
        float4 s = make_float4(0.f, 0.f, 0.f, 0.f);
#pragma unroll
        for (int w2 = 0; w2 < 8; ++w2) {
            float4 p = s_part[w2][i];
            s.x += p.x; s.y += p.y; s.z += p.z; s.w += p.w;
        }
        vs4[i] = make_float4(s.x * inv, s.y * inv, s.z * inv, s.w * inv);
    }
}

// ---------------------------------------------------------------------------
// Kq: q[b] = qx[b,:] . w_mlp[D:]   (one wave per b)
// ---------------------------------------------------------------------------
__global__ void qdot_kernel(const float* __restrict__ qx,
                            const float* __restrict__ wmlp,
                            float* __restrict__ q) {
    int b = blockIdx.x;
    int lane = threadIdx.x;
    float s = 0.0f;
    for (int d = lane; d < DD; d += 32) s += qx[b * DD + d] * wmlp[DD + d];
    for (int off = 16; off > 0; off >>= 1) s += __shfl_xor(s, off, 32);
    if (lane == 0) q[b] = s;
}

// ---------------------------------------------------------------------------
// K2: score = softmax_l(tanh(u + q)), m[b,:] = sum_l score_l * w_l * embed_row
// (float4 gather; embed table is L2-resident on the second pass)
// ---------------------------------------------------------------------------
__global__ void pass2_kernel(const int* __restrict__ text,
                             const float* __restrict__ embed,
                             const float* __restrict__ lens,
                             const float* __restrict__ u,
                             const float* __restrict__ q,
                             float* __restrict__ m_out) {
    __shared__ int    s_text[LL_TXT];
    __shared__ float  s_score[LL_TXT];
    __shared__ float  s_red[256];
    __shared__ float4 s_part[8][DD4 + 1];
    int b = blockIdx.x;
    int tid = threadIdx.x;
    int wave = tid >> 5, lane = tid & 31;

    float qb = q[b];
    float start = lens[b * 4 + 0];
    float endv  = lens[b * 4 + 1];
    float mlen  = lens[b * 4 + 2];
    for (int i = tid; i < LL_TXT; i += 256) s_text[i] = text[b * LL_TXT + i];

    float e0 = tanhf(u[b * LL_TXT + tid] + qb);
    float e1 = tanhf(u[b * LL_TXT + tid + 256] + qb);
    s_red[tid] = fmaxf(e0, e1);
    __syncthreads();
    for (int s = 128; s > 0; s >>= 1) {
        if (tid < s) s_red[tid] = fmaxf(s_red[tid], s_red[tid + s]);
        __syncthreads();
    }
    float mx = s_red[0];
    __syncthreads();
    float x0 = expf(e0 - mx), x1 = expf(e1 - mx);
    s_red[tid] = x0 + x1;
    __syncthreads();
    for (int s = 128; s > 0; s >>= 1) {
        if (tid < s) s_red[tid] += s_red[tid + s];
        __syncthreads();
    }
    float inv = 1.0f / s_red[0];
    s_score[tid] = x0 * inv;
    s_score[tid + 256] = x1 * inv;
    __syncthreads();

    float4 acc[3];
#pragma unroll
    for (int t = 0; t < 3; ++t) acc[t] = make_float4(0.f, 0.f, 0.f, 0.f);

    for (int l = wave; l < LL_TXT; l += 8) {
        float fl = (float)l;
        float ldst = fl < start ? (start - fl) : (fl <= endv ? 0.0f : fl - endv);
        float w = 1.0f - ldst / mlen;
        w = (fl < mlen) ? w : 0.0f;
        float coef = s_score[l] * w;          // wave-uniform
        if (coef != 0.0f) {
            const float4* row4 = (const float4*)(embed + (long)s_text[l] * DD);
#pragma unroll
            for (int t = 0; t < 3; ++t) {
                int i = lane + 32 * t;
                if (i < DD4) {
                    float4 v = row4[i];
                    acc[t].x += coef * v.x; acc[t].y += coef * v.y;
                    acc[t].z += coef * v.z; acc[t].w += coef * v.w;
                }
            }
        }
    }
#pragma unroll
    for (int t = 0; t < 3; ++t) {
        int i = lane + 32 * t;
        if (i < DD4) s_part[wave][i] = acc[t];
    }
    __syncthreads();
    float4* m4 = (float4*)(m_out + (size_t)b * DD);
    for (int i = tid; i < DD4; i += 256) {
        float4 s = make_float4(0.f, 0.f, 0.f, 0.f);
#pragma unroll
        for (int w2 = 0; w2 < 8; ++w2) {
            float4 p = s_part[w2][i];
            s.x += p.x; s.y += p.y; s.z += p.z; s.w += p.w;
        }
        m4[i] = s;
    }
}

// ---------------------------------------------------------------------------
// WMMA GEMM: out[M,N] = act(A[M,K] @ W[K,N] + bias + add)
// one wave32 per 16x16 tile, V_WMMA_F32_16X16X4_F32, K mult of 4, M mult of 16.
// Edge columns handled with clamped addresses + branch-free v_cndmask selects
// (no exec-divergent loads in the inner loop).
// A 16x4 layout: lanes 0-15 hold K={0,1}, lanes 16-31 hold K={2,3}
// C/D 16x16   : VGPR r -> rows r (lanes 0-15) and r+8 (lanes 16-31)
// ---------------------------------------------------------------------------
__global__ void gemm_wmma_kernel(const float* __restrict__ A,
                                 const float* __restrict__ W,
                                 const float* __restrict__ bias,
                                 const float* __restrict__ add,
                                 float* __restrict__ out,
                                 int M, int K, int N, int act) {
    int lane  = threadIdx.x;      // 0..31
    int tn    = blockIdx.x * 16;  // tile col base
    int tm    = blockIdx.y * 16;  // tile row base
    int half  = lane >> 4;        // 0 or 1
    int idx16 = lane & 15;

    int m = tm + idx16;
    int n = tn + idx16;
    bool nok = (n < N);
    int nc = nok ? n : (N - 1);          // clamped: always a valid address
    const float* Arow = A + (size_t)m * K;
    const float* Wcol = W + nc;

    v8f acc = {};
    for (int k0 = 0; k0 < K; k0 += 4) {
        int ka = k0 + half * 2;
        v2f a, bvec;
        a.x = Arow[ka];
        a.y = Arow[ka + 1];
        float w0 = Wcol[(size_t)ka * N];
        float w1 = Wcol[(size_t)(ka + 1) * N];
        bvec.x = nok ? w0 : 0.0f;        // branch-free select
        bvec.y = nok ? w1 : 0.0f;
        acc = __builtin_amdgcn_wmma_f32_16x16x4_f32(false, a, false, bvec,
                                                    (short)0, acc, false, false);
    }

    if (nok) {
        float bb = bias ? bias[n] : 0.0f;
        int rbase = tm + half * 8;
#pragma unroll
        for (int r = 0; r < 8; ++r) {
            float v = acc[r] + bb;
            int row = rbase + r;
            if (add) v += add[(size_t)row * N + n];
            if (act) v = tanhf(v);
            out[(size_t)row * N + n] = v;
        }
    }
}

// ---------------------------------------------------------------------------
// final 3-way softmax
// ---------------------------------------------------------------------------
__global__ void softmax3_kernel(const float* __restrict__ logits,
                                float* __restrict__ out) {
    int b = blockIdx.x * blockDim.x + threadIdx.x;
    if (b < BB) {
        float a = logits[b * 3 + 0];
        float c = logits[b * 3 + 1];
        float d = logits[b * 3 + 2];
        float mx = fmaxf(a, fmaxf(c, d));
        float ea = expf(a - mx), ec = expf(c - mx), ed = expf(d - mx);
        float inv = 1.0f / (ea + ec + ed);
        out[b * 3 + 0] = ea * inv;
        out[b * 3 + 1] = ec * inv;
        out[b * 3 + 2] = ed * inv;
    }
}

// ---------------------------------------------------------------------------
extern "C" void kernel_launch(void* const* d_in, const int* in_sizes, int n_in,
                              void* d_out, int out_size, void* d_ws, size_t ws_size,
                              hipStream_t stream) {
    const int*   text   = (const int*)d_in[0];
    const int*   aspect = (const int*)d_in[1];
    const int*   left   = (const int*)d_in[2];
    const float* embed  = (const float*)d_in[3];
    const float* Wx     = (const float*)d_in[4];
    // d_in[5] = Ws : dead code in the reference (s is never used)
    const float* Wk     = (const float*)d_in[6];
    const float* bk     = (const float*)d_in[7];
    const float* Wq     = (const float*)d_in[8];
    const float* bq     = (const float*)d_in[9];
    const float* wmlp   = (const float*)d_in[10];
    const float* Wproj  = (const float*)d_in[11];
    const float* bproj  = (const float*)d_in[12];
    const float* Wm     = (const float*)d_in[13];
    const float* bm     = (const float*)d_in[14];
    const float* Wd     = (const float*)d_in[15];
    const float* bd     = (const float*)d_in[16];

    float* ws  = (float*)d_ws;
    float* out = (float*)d_out;

    const size_t BD = (size_t)BB * DD;   // 76800 floats (16B-aligned stride)
    size_t o = 0;
    float* asp_e   = ws + o; o += BD;
    float* v_s     = ws + o; o += BD;
    float* u       = ws + o; o += (size_t)BB * LL_TXT;
    float* m_buf   = ws + o; o += BD;
    float* tA      = ws + o; o += BD;
    float* tB      = ws + o; o += BD;
    float* qx      = ws + o; o += BD;
    float* q       = ws + o; o += BB;
    float* cvec_d0 = ws + o; o += (DD + 4);
    float* lens    = ws + o; o += (size_t)BB * 4;
    float* logits  = ws + o; o += (size_t)BB * PP;
    (void)o; (void)ws_size; (void)n_in; (void)in_sizes; (void)out_size;

    // stage 0: lengths, aspect mean, cvec
    prep_kernel<<<BB, 256, 0, stream>>>(text, aspect, left, embed, asp_e, lens);
    cvec_kernel<<<2, 256, 0, stream>>>(Wk, bk, wmlp, cvec_d0);

    // stage 1: gather pass -> u (pre-softmax row scores) and v_s
    pass1_kernel<<<BB, 256, 0, stream>>>(text, embed, cvec_d0, lens, u, v_s);

    // stage 2: x3 = asp_e @ Wx^3 ; qx = x3 @ Wq + bq ; q = qx . w_mlp[D:]
    dim3 gD((DD + 15) / 16, BB / 16);
    gemm_wmma_kernel<<<gD, 32, 0, stream>>>(asp_e, Wx, nullptr, nullptr, tA, BB, DD, DD, 0);
    gemm_wmma_kernel<<<gD, 32, 0, stream>>>(tA, Wx, nullptr, nullptr, tB, BB, DD, DD, 0);
    gemm_wmma_kernel<<<gD, 32, 0, stream>>>(tB, Wx, nullptr, nullptr, tA, BB, DD, DD, 0);
    gemm_wmma_kernel<<<gD, 32, 0, stream>>>(tA, Wq, bq, nullptr, qx, BB, DD, DD, 0);
    qdot_kernel<<<BB, 32, 0, stream>>>(qx, wmlp, q);

    // stage 3: softmax over L + score-weighted gather -> m
    pass2_kernel<<<BB, 256, 0, stream>>>(text, embed, lens, u, q, m_buf);

    // stage 4: v_ts = (m@Wk+bk)@Wproj+bproj ; v_ns = v_ts+v_s ;
    //          v_ms = tanh(v_ns@Wm+bm) ; logits = v_ms@Wd+bd ; softmax
    gemm_wmma_kernel<<<gD, 32, 0, stream>>>(m_buf, Wk, bk, nullptr, tA, BB, DD, DD, 0);
    gemm_wmma_kernel<<<gD, 32, 0, stream>>>(tA, Wproj, bproj, v_s, tB, BB, DD, DD, 0);
    gemm_wmma_kernel<<<gD, 32, 0, stream>>>(tB, Wm, bm, nullptr, tA, BB, DD, DD, 1);
    dim3 gP(1, BB / 16);
    gemm_wmma_kernel<<<gP, 32, 0, stream>>>(tA, Wd, bd, nullptr, logits, BB, DD, PP, 0);
    softmax3_kernel<<<1, 256, 0, stream>>>(logits, out);
}